// TokenEmbedding_7413113553153
// MI455X (gfx1250) — compile-verified
//
#include <hip/hip_runtime.h>
#include <math.h>

// Problem constants (from the reference)
#define VOCAB    50257
#define D_MODEL  1024
#define BATCH    4
#define SEQ      4096
#define ROWS     (BATCH * SEQ)          // 16384 output rows
#define WAVE     32                     // gfx1250 is wave32-only
#define WAVES_PB 8                      // 256-thread block = 8 waves
#define F4_ROW   (D_MODEL / 4)          // 256 float4 per row
#define F4_LANE  (F4_ROW / WAVE)        // 8 float4 per lane

// native clang vector: legal operand for __builtin_nontemporal_store and
// guarantees b128 vmem ops
typedef float v4f __attribute__((ext_vector_type(4)));

// out[row, :] = emb[tok[row], :] * sqrt(D_MODEL) + PE(row % SEQ, :)
// One wave per row: lane L handles float4 chunks {L, L+32, ..., L+224}
// -> fully coalesced 4KB row reads (global_load_b128) and NT row writes.
__global__ __launch_bounds__(256) void tokemb_pe_kernel(
    const int*   __restrict__ tok,   // [ROWS]
    const float* __restrict__ emb,   // [VOCAB, D_MODEL]
    float*       __restrict__ out)   // [ROWS, D_MODEL]
{
    const int lane = threadIdx.x & (WAVE - 1);
    const int wv   = threadIdx.x >> 5;               // wave id in block
    const int row  = blockIdx.x * WAVES_PB + wv;     // 0..ROWS-1
    if (row >= ROWS) return;

    const int s = row & (SEQ - 1);                   // sequence position
    // wave-uniform token index (one dword per wave; L2/RT cached)
    const int t = tok[row];

    const v4f* __restrict__ src =
        (const v4f*)(emb + (long long)t * D_MODEL);
    v4f* __restrict__ dst =
        (v4f*)(out + (long long)row * D_MODEL);

    const float sp = (float)s;
    // inv_freq(k) = 10000^(-k/512) = 2^(k*c),  c = -log2(10000)/512
    const float c  = -13.2877123795494f / 512.0f;
    const float R  = 0.9821719f;                     // 2^c : step k -> k+1

    #pragma unroll
    for (int i = 0; i < F4_LANE; ++i) {
        const int f4 = lane + WAVE * i;              // float4 index in row
        // gather read: default temporal hint — token reuse means these
        // rows live in the 192MB L2 (table is ~206MB, nearly resident)
        const v4f v = src[f4];

        // PE for dims 4*f4 .. 4*f4+3  == pairs k0, k0+1
        const int   k0 = 2 * f4;
        const float a0 = sp * exp2f((float)k0 * c);  // angle for pair k0
        const float a1 = a0 * R;                     // angle for pair k0+1
        float s0, c0, s1, c1;
        __sincosf(a0, &s0, &c0);
        __sincosf(a1, &s1, &c1);

        v4f r;
        r.x = fmaf(v.x, 32.0f, s0);                  // sqrt(1024) = 32
        r.y = fmaf(v.y, 32.0f, c0);
        r.z = fmaf(v.z, 32.0f, s1);
        r.w = fmaf(v.w, 32.0f, c1);

        // streamed output: non-temporal store (TH_NT) -> don't evict the
        // embedding table from L2 with write-once data
        __builtin_nontemporal_store(r, &dst[f4]);
    }
}

extern "C" void kernel_launch(void* const* d_in, const int* in_sizes, int n_in,
                              void* d_out, int out_size, void* d_ws, size_t ws_size,
                              hipStream_t stream) {
    const int*   tok = (const int*)d_in[0];    // input_x [BATCH, SEQ] int32
    const float* emb = (const float*)d_in[1];  // emb_table [VOCAB, D_MODEL] f32
    float*       out = (float*)d_out;          // [BATCH, SEQ, D_MODEL] f32

    (void)in_sizes; (void)n_in; (void)out_size; (void)d_ws; (void)ws_size;

    const int blocks = ROWS / WAVES_PB;        // 2048, exact
    tokemb_pe_kernel<<<blocks, WAVES_PB * WAVE, 0, stream>>>(tok, emb, out);
}